// SHT_75660143886823
// MI455X (gfx1250) — compile-verified
//
#include <hip/hip_runtime.h>
#include <math.h>

// ---------------- problem constants (reference: NSIDE = 128) ----------------
#define NSIDE 128
#define NLAT  (4*NSIDE - 1)      // 511 rings
#define NLON  (4*NSIDE)          // 512 max pixels per ring
#define LMAX  NLAT               // 511
#define MMAX  (NLON/2 + 1)       // 257
#define NPIX  (12*NSIDE*NSIDE)   // 196608
#define KPAD  512                // nlat padded to multiple of 16 for WMMA K loop

#define TWO_PI_F 6.2831853071795864769f

typedef __attribute__((ext_vector_type(2))) float v2f;
typedef __attribute__((ext_vector_type(8))) float v8f;

// ============================================================================
// Kernel 1: per-ring real DFT + CS phase  ->  ftm_r/ftm_i [KPAD][MMAX] in ws.
// One block per ring. Twiddles exp(2pi k/nphi) tabulated once in LDS
// (<=512 sincosf per ring), inner loop indexes table with incremental
// (m*p mod nphi) -- valid because active m <= nphi/2 < nphi.
// ============================================================================
__global__ __launch_bounds__(256) void sht_ring_dft(
    const float* __restrict__ x,
    float* __restrict__ ftm_r,
    float* __restrict__ ftm_i)
{
    __shared__ float fring[NLON];
    __shared__ float twc[NLON];
    __shared__ float tws[NLON];

    const int t   = blockIdx.x;
    const int tid = threadIdx.x;

    if (t >= NLAT) {                       // K padding ring -> zeros
        for (int m = tid; m < MMAX; m += 256) {
            ftm_r[t*MMAX + m] = 0.0f;
            ftm_i[t*MMAX + m] = 0.0f;
        }
        return;
    }

    // ---- HEALPix ring geometry (closed forms) ----
    int nphi, offs;
    float shift;
    if (t < NSIDE - 1) {                    // north polar cap
        nphi  = 4*(t + 1);
        offs  = 2*t*(t + 1);
        shift = 0.5f;
    } else if (t < 3*NSIDE) {               // equatorial belt
        nphi  = 4*NSIDE;
        offs  = 2*(NSIDE - 1)*NSIDE + (t - (NSIDE - 1))*4*NSIDE;
        shift = 0.5f * (float)((t - NSIDE + 2) & 1);
    } else {                                // south polar cap
        const int r = 4*NSIDE - 1 - t;
        nphi  = 4*r;
        offs  = NPIX - 2*r*(r + 1);
        shift = 0.5f;
    }
    const float phi0 = (TWO_PI_F / (float)nphi) * shift;

    // ---- stage ring pixels + twiddle table into LDS ----
    for (int p = tid; p < NLON; p += 256) {
        fring[p] = (p < nphi) ? x[offs + p] : 0.0f;
        if (p < nphi) {
            float s, c;
            sincosf((TWO_PI_F * (float)p) / (float)nphi, &s, &c);
            twc[p] = c;
            tws[p] = s;
        } else {
            twc[p] = 0.0f;
            tws[p] = 0.0f;
        }
    }
    __syncthreads();

    const int mlim = min(MMAX, nphi/2 + 1);    // mmask
    for (int m = tid; m < MMAX; m += 256) {
        float outr = 0.0f, outi = 0.0f;
        if (m < mlim) {
            float ar = 0.0f, ai = 0.0f;
            int idx = 0;                        // (m*p) mod nphi, incremental
            for (int p = 0; p < nphi; ++p) {
                const float f = fring[p];
                ar = fmaf(f, twc[idx], ar);
                ai = fmaf(f, tws[idx], ai);
                idx += m;
                if (idx >= nphi) idx -= nphi;
            }
            // W = exp(-2pi i m p / nphi): real = +cos sum, imag = -sin sum
            const float fr = ar, fi = -ai;
            float ps, pc;                       // phase = exp(-i m phi0)
            sincosf((float)m * phi0, &ps, &pc);
            outr = fr*pc + fi*ps;
            outi = fi*pc - fr*ps;
        }
        ftm_r[t*MMAX + m] = outr;
        ftm_i[t*MMAX + m] = outi;
    }
}

// ============================================================================
// Kernel 2: out[l,m] = sum_k weights[m,l,k] * ftm[k,m]   (re & im together)
// Bandwidth bound: 268 MB of f32 weights read exactly once (~11.5 us floor
// at 23.3 TB/s). One wave per (m, l-tile of 16); V_WMMA_F32_16X16X4_F32
// accumulates D(16x16) with N=0 -> real, N=1 -> imag, N>=2 -> zero columns.
//   A (16x4):  lane M = lane&15; VGPR0 = K(2*(lane>>4)), VGPR1 = K+1
//   B (4x16):  lane N = lane&15; VGPR0 = K(2*(lane>>4)), VGPR1 = K+1
//   D (16x16): lane N = lane&15; VGPR j = M = j + 8*(lane>>4)
// EXEC stays all-ones through the WMMA loop (masking is branch-free).
// ============================================================================
#define LTILES 32                       // ceil(511/16)
#define WPB    8                        // waves per 256-thread block

__global__ __launch_bounds__(256) void sht_legendre_wmma(
    const float* __restrict__ w,        // [MMAX][LMAX][NLAT]
    const float* __restrict__ ftm_r,    // [KPAD][MMAX]
    const float* __restrict__ ftm_i,    // [KPAD][MMAX]
    float* __restrict__ out)            // [LMAX][MMAX] complex interleaved
{
    const int lane = threadIdx.x & 31;
    const int wid  = blockIdx.x * WPB + (threadIdx.x >> 5);
    const int m    = wid >> 5;          // 0..256
    const int lt   = wid & 31;          // 0..31

    const int rowc = lane & 15;         // A row (M) / B+D column (N)
    const int h    = lane >> 4;         // K pair selector

    const int  l    = lt*16 + rowc;
    const long lrow = (long)min(l, LMAX - 1);     // clamp so loads stay in-bounds
    const float* __restrict__ wrow =
        w + (long)m * (long)(LMAX * NLAT) + lrow * (long)NLAT;

    // B source per lane: col 0 = real, col 1 = imag, cols >= 2 masked to zero
    const float* __restrict__ bsrc = (rowc == 1) ? ftm_i : ftm_r;
    const float bmask = (rowc < 2) ? 1.0f : 0.0f;

    v8f c = {0.f, 0.f, 0.f, 0.f, 0.f, 0.f, 0.f, 0.f};

    for (int k = 0; k < KPAD; k += 16) {
        __builtin_prefetch(wrow + k + 128, 0, 0);   // global_prefetch_b8
#pragma unroll
        for (int kk = 0; kk < 16; kk += 4) {
            const int kb = k + kk;
            const int k0 = kb + 2*h;                // <= 510, always valid
            const int k1 = k0 + 1;                  // == 511 only at the tail
            v2f a, b;
            a.x = wrow[k0];
            {   // clamped-address load + select (no branch, no OOB)
                const float t1 = wrow[min(k1, NLAT - 1)];
                a.y = (k1 < NLAT) ? t1 : 0.0f;
            }
            b.x = bmask * bsrc[(size_t)k0 * MMAX + m];   // k1 <= 511 < KPAD
            b.y = bmask * bsrc[(size_t)k1 * MMAX + m];
            c = __builtin_amdgcn_wmma_f32_16x16x4_f32(
                    /*neg_a=*/false, a, /*neg_b=*/false, b,
                    /*c_mod=*/(short)0, c, /*reuse_a=*/false, /*reuse_b=*/false);
        }
    }

    // ---- write D: lanes N==0 -> real, N==1 -> imag ----
    if (rowc < 2) {
#pragma unroll
        for (int j = 0; j < 8; ++j) {
            const int ll = lt*16 + 8*h + j;
            if (ll < LMAX)
                out[((size_t)ll * MMAX + m) * 2 + rowc] = c[j];
        }
    }
}

// ============================================================================
extern "C" void kernel_launch(void* const* d_in, const int* in_sizes, int n_in,
                              void* d_out, int out_size, void* d_ws, size_t ws_size,
                              hipStream_t stream) {
    (void)in_sizes; (void)n_in; (void)out_size; (void)ws_size;
    const float* x = (const float*)d_in[0];          // [NPIX]
    const float* w = (const float*)d_in[1];          // [MMAX][LMAX][NLAT]
    float* out = (float*)d_out;                      // [LMAX][MMAX] interleaved c64

    float* ftm_r = (float*)d_ws;                     // [KPAD][MMAX]
    float* ftm_i = ftm_r + (size_t)KPAD * MMAX;      // [KPAD][MMAX]

    // Stage 1: 512 blocks (511 rings + 1 zero pad ring)
    sht_ring_dft<<<KPAD, 256, 0, stream>>>(x, ftm_r, ftm_i);

    // Stage 2: 257 m-values * 32 l-tiles = 8224 waves, 8 waves/block
    const int nwaves = MMAX * LTILES;                // 8224
    sht_legendre_wmma<<<nwaves / WPB, 256, 0, stream>>>(w, ftm_r, ftm_i, out);
}